// GCN_78245714199374
// MI455X (gfx1250) — compile-verified
//
#include <hip/hip_runtime.h>

// ---------------------------------------------------------------------------
// PDNConv GNN forward for MI455X (gfx1250, wave32).
// All dense matmuls use V_WMMA_F32_16X16X4_F32 (reference dtype is f32).
// ---------------------------------------------------------------------------

#define N_NODES 50000
#define N_EDGES 800000
#define EDGE_F  16
#define HIDDEN  64
#define CLASSES 10
#define N_PICK  4096

typedef float v2f __attribute__((ext_vector_type(2)));
typedef float v8f __attribute__((ext_vector_type(8)));

static __device__ __forceinline__ v8f wmma_f32(v2f a, v2f b, v8f c) {
  // D = A(16x4 f32) * B(4x16 f32) + C(16x16 f32)
  return __builtin_amdgcn_wmma_f32_16x16x4_f32(
      /*neg_a=*/false, a, /*neg_b=*/false, b,
      /*c_mod=*/(short)0, c, /*reuse_a=*/false, /*reuse_b=*/false);
}

static __device__ __forceinline__ v8f splat8(float x) {
  v8f r;
#pragma unroll
  for (int i = 0; i < 8; ++i) r[i] = x;
  return r;
}

// ---------------------------------------------------------------------------
// Kernel 1: edge encoder + per-layer gating weights.
//   ea = (edge_attr @ encW1 + b1) @ encW2 + b2                  [E,16]
//   ew[l][e] = sigmoid( relu(ea @ mlpW1[l] + mlpb1[l]) . mlpW2[l] + mlpb2[l] )
// One wave handles a 16-edge tile; all GEMMs via WMMA f32 16x16x4.
// ---------------------------------------------------------------------------
__global__ __launch_bounds__(256) void edge_weights_kernel(
    const float* __restrict__ edge_attr,
    const float* __restrict__ encW1, const float* __restrict__ encb1,
    const float* __restrict__ encW2, const float* __restrict__ encb2,
    const float* __restrict__ mlpW1, const float* __restrict__ mlpb1,
    const float* __restrict__ mlpW2, const float* __restrict__ mlpb2,
    float* __restrict__ ew) {
  __shared__ float sW1[16 * 16], sb1[16], sW2[16 * 16], sb2[16];
  __shared__ float sM1[4 * 16 * 64], sMb1[4 * 64], sM2[4 * 64], sMb2[4];
  __shared__ float sEA[8][16 * 16];  // per-wave ea tile scratch
  __shared__ float sT[8][16 * 16];   // per-wave gating-hidden scratch

  for (int i = threadIdx.x; i < 256; i += 256)  sW1[i] = encW1[i];
  for (int i = threadIdx.x; i < 16;  i += 256)  sb1[i] = encb1[i];
  for (int i = threadIdx.x; i < 256; i += 256)  sW2[i] = encW2[i];
  for (int i = threadIdx.x; i < 16;  i += 256)  sb2[i] = encb2[i];
  for (int i = threadIdx.x; i < 4096; i += 256) sM1[i] = mlpW1[i];
  for (int i = threadIdx.x; i < 256; i += 256)  sMb1[i] = mlpb1[i];
  for (int i = threadIdx.x; i < 256; i += 256)  sM2[i] = mlpW2[i];
  for (int i = threadIdx.x; i < 4;   i += 256)  sMb2[i] = mlpb2[i];
  __syncthreads();

  const int lane = threadIdx.x & 31;
  const int wid  = threadIdx.x >> 5;
  const int m16  = lane & 15;   // A: row M / B,C: col N
  const int half = lane >> 4;
  const int gw = blockIdx.x * 8 + wid;
  const int nw = gridDim.x * 8;
  const int NT = N_EDGES / 16;  // 50000 exactly
  float* eat = sEA[wid];
  float* tt  = sT[wid];

  for (int t = gw; t < NT; t += nw) {
    const int e0 = t * 16;
    v2f a[4];
    // A fragments of raw edge_attr tile (16 edges x 16 feat, K split in 4)
#pragma unroll
    for (int kk = 0; kk < 4; ++kk) {
      const int k0 = 4 * kk + 2 * half;
      const float* p = edge_attr + (size_t)(e0 + m16) * EDGE_F + k0;
      a[kk].x = p[0]; a[kk].y = p[1];
    }
    // ea1 = A @ W1 + b1
    v8f c = splat8(sb1[m16]);
#pragma unroll
    for (int kk = 0; kk < 4; ++kk) {
      const int k0 = 4 * kk + 2 * half;
      v2f b; b.x = sW1[k0 * 16 + m16]; b.y = sW1[(k0 + 1) * 16 + m16];
      c = wmma_f32(a[kk], b, c);
    }
#pragma unroll
    for (int v = 0; v < 8; ++v) eat[(v + 8 * half) * 16 + m16] = c[v];
    __builtin_amdgcn_wave_barrier();
    // reload ea1 as A fragments
#pragma unroll
    for (int kk = 0; kk < 4; ++kk) {
      const int k0 = 4 * kk + 2 * half;
      a[kk].x = eat[m16 * 16 + k0]; a[kk].y = eat[m16 * 16 + k0 + 1];
    }
    __builtin_amdgcn_wave_barrier();
    // ea2 = ea1 @ W2 + b2
    c = splat8(sb2[m16]);
#pragma unroll
    for (int kk = 0; kk < 4; ++kk) {
      const int k0 = 4 * kk + 2 * half;
      v2f b; b.x = sW2[k0 * 16 + m16]; b.y = sW2[(k0 + 1) * 16 + m16];
      c = wmma_f32(a[kk], b, c);
    }
#pragma unroll
    for (int v = 0; v < 8; ++v) eat[(v + 8 * half) * 16 + m16] = c[v];
    __builtin_amdgcn_wave_barrier();
#pragma unroll
    for (int kk = 0; kk < 4; ++kk) {
      const int k0 = 4 * kk + 2 * half;
      a[kk].x = eat[m16 * 16 + k0]; a[kk].y = eat[m16 * 16 + k0 + 1];
    }

    // gating MLP per layer: relu(ea @ mlpW1[l] + b) . mlpW2[l] + b2 -> sigmoid
#pragma unroll
    for (int l = 0; l < 4; ++l) {
      float acc = 0.0f;
#pragma unroll
      for (int j = 0; j < 4; ++j) {        // 4 N-tiles of the 64-wide hidden
        const int nb = 16 * j;
        v8f tc = splat8(sMb1[l * 64 + nb + m16]);
#pragma unroll
        for (int kk = 0; kk < 4; ++kk) {
          const int k0 = 4 * kk + 2 * half;
          v2f b;
          b.x = sM1[l * 1024 + k0 * 64 + nb + m16];
          b.y = sM1[l * 1024 + (k0 + 1) * 64 + nb + m16];
          tc = wmma_f32(a[kk], b, tc);
        }
#pragma unroll
        for (int v = 0; v < 8; ++v) {
          float r = tc[v];
          tt[(v + 8 * half) * 16 + m16] = r > 0.0f ? r : 0.0f;
        }
        __builtin_amdgcn_wave_barrier();
        // dot with mlpW2 slice; both half-waves compute (dup), lanes 0-15 store
        float p = 0.0f;
#pragma unroll
        for (int n = 0; n < 16; ++n) p += tt[m16 * 16 + n] * sM2[l * 64 + nb + n];
        acc += p;
        __builtin_amdgcn_wave_barrier();
      }
      if (half == 0) {
        const float z = acc + sMb2[l];
        ew[(size_t)l * N_EDGES + e0 + m16] = 1.0f / (1.0f + __expf(-z));
      }
    }
  }
}

// ---------------------------------------------------------------------------
// Degree / gcn_norm kernels (self loop contributes 1.0 -> init deg = 1).
// ---------------------------------------------------------------------------
__global__ void deg_init_kernel(float* __restrict__ deg4) {
  int i = blockIdx.x * blockDim.x + threadIdx.x;
  if (i < 4 * N_NODES) deg4[i] = 1.0f;
}

__global__ void deg_accum_kernel(const int* __restrict__ col,
                                 const float* __restrict__ ew,
                                 float* __restrict__ deg4) {
  int e = blockIdx.x * blockDim.x + threadIdx.x;
  if (e < N_EDGES) {
    int c = col[e];
#pragma unroll
    for (int l = 0; l < 4; ++l)
      atomicAdd(&deg4[(size_t)l * N_NODES + c], ew[(size_t)l * N_EDGES + e]);
  }
}

__global__ void deg_rsqrt_kernel(float* __restrict__ deg4) {
  int i = blockIdx.x * blockDim.x + threadIdx.x;
  if (i < 4 * N_NODES) {
    float d = deg4[i];
    deg4[i] = d > 0.0f ? rsqrtf(d) : 0.0f;
  }
}

// ---------------------------------------------------------------------------
// Kernel: h = (relu?)(xin) @ W ; xout = bias + dis^2 * h  (self-loop fused)
// One wave per 16-row tile, 4 N-tiles x 16 k-steps of WMMA f32 16x16x4.
// ---------------------------------------------------------------------------
__global__ __launch_bounds__(256) void conv_gemm_kernel(
    const float* __restrict__ xin, const float* __restrict__ W,
    const float* __restrict__ bias, const float* __restrict__ dis,
    float* __restrict__ h, float* __restrict__ xout, int relu_in) {
  __shared__ float sW[64 * 64];
  __shared__ float sB[64];
  for (int i = threadIdx.x; i < 4096; i += 256) sW[i] = W[i];
  if (threadIdx.x < 64) sB[threadIdx.x] = bias[threadIdx.x];
  __syncthreads();

  const int lane = threadIdx.x & 31;
  const int wid  = threadIdx.x >> 5;
  const int m16  = lane & 15;
  const int half = lane >> 4;
  const int gw = blockIdx.x * 8 + wid;
  const int nw = gridDim.x * 8;
  const int NT = N_NODES / 16;  // 3125 exactly

  for (int t = gw; t < NT; t += nw) {
    const int r0 = t * 16;
    v8f c0 = splat8(0.0f), c1 = splat8(0.0f), c2 = splat8(0.0f), c3 = splat8(0.0f);
#pragma unroll
    for (int kk = 0; kk < 16; ++kk) {
      const int k0 = 4 * kk + 2 * half;
      const float* p = xin + (size_t)(r0 + m16) * HIDDEN + k0;
      v2f a; a.x = p[0]; a.y = p[1];
      if (relu_in) {
        a.x = a.x > 0.0f ? a.x : 0.0f;
        a.y = a.y > 0.0f ? a.y : 0.0f;
      }
      v2f b;
      b.x = sW[k0 * 64 + 0  + m16]; b.y = sW[(k0 + 1) * 64 + 0  + m16]; c0 = wmma_f32(a, b, c0);
      b.x = sW[k0 * 64 + 16 + m16]; b.y = sW[(k0 + 1) * 64 + 16 + m16]; c1 = wmma_f32(a, b, c1);
      b.x = sW[k0 * 64 + 32 + m16]; b.y = sW[(k0 + 1) * 64 + 32 + m16]; c2 = wmma_f32(a, b, c2);
      b.x = sW[k0 * 64 + 48 + m16]; b.y = sW[(k0 + 1) * 64 + 48 + m16]; c3 = wmma_f32(a, b, c3);
    }
#pragma unroll
    for (int v = 0; v < 8; ++v) {
      const int m = r0 + v + 8 * half;
      const float d = dis[m];
      const float dd = d * d;
      float* hr = h + (size_t)m * HIDDEN;
      float* xr = xout + (size_t)m * HIDDEN;
      hr[0  + m16] = c0[v]; xr[0  + m16] = sB[0  + m16] + dd * c0[v];
      hr[16 + m16] = c1[v]; xr[16 + m16] = sB[16 + m16] + dd * c1[v];
      hr[32 + m16] = c2[v]; xr[32 + m16] = sB[32 + m16] + dd * c2[v];
      hr[48 + m16] = c3[v]; xr[48 + m16] = sB[48 + m16] + dd * c3[v];
    }
  }
}

// ---------------------------------------------------------------------------
// Scatter: xout[col] += (dis[row]*ew*dis[col]) * h[row]; 64 lanes per edge.
// ---------------------------------------------------------------------------
__global__ __launch_bounds__(256) void scatter_kernel(
    const int* __restrict__ row, const int* __restrict__ col,
    const float* __restrict__ ew_l, const float* __restrict__ dis,
    const float* __restrict__ h, float* __restrict__ xout) {
  const long long gid = (long long)blockIdx.x * blockDim.x + threadIdx.x;
  if (gid >= (long long)N_EDGES * HIDDEN) return;
  const int e = (int)(gid >> 6);
  const int f = (int)(gid & 63);
  const int r = row[e];
  const int c = col[e];
  const float nrm = dis[r] * ew_l[e] * dis[c];
  atomicAdd(&xout[(size_t)c * HIDDEN + f], nrm * h[(size_t)r * HIDDEN + f]);
}

// ---------------------------------------------------------------------------
// Classifier: logits = x[pick] @ linW + linb ; softmax over 10 classes.
// ---------------------------------------------------------------------------
__global__ __launch_bounds__(256) void classify_kernel(
    const float* __restrict__ x, const float* __restrict__ linW,
    const float* __restrict__ linb, const int* __restrict__ pick,
    float* __restrict__ out) {
  const int i = blockIdx.x * blockDim.x + threadIdx.x;
  if (i >= N_PICK) return;
  const int p = pick[i];
  const float* xp = x + (size_t)p * HIDDEN;
  float logit[CLASSES];
#pragma unroll
  for (int c = 0; c < CLASSES; ++c) logit[c] = linb[c];
  for (int hh = 0; hh < HIDDEN; ++hh) {
    const float xv = xp[hh];
#pragma unroll
    for (int c = 0; c < CLASSES; ++c) logit[c] += xv * linW[hh * CLASSES + c];
  }
  float mx = logit[0];
#pragma unroll
  for (int c = 1; c < CLASSES; ++c) mx = logit[c] > mx ? logit[c] : mx;
  float s = 0.0f;
#pragma unroll
  for (int c = 0; c < CLASSES; ++c) { float e = __expf(logit[c] - mx); logit[c] = e; s += e; }
  const float inv = 1.0f / s;
#pragma unroll
  for (int c = 0; c < CLASSES; ++c) out[(size_t)i * CLASSES + c] = logit[c] * inv;
}

// ---------------------------------------------------------------------------
extern "C" void kernel_launch(void* const* d_in, const int* in_sizes, int n_in,
                              void* d_out, int out_size, void* d_ws, size_t ws_size,
                              hipStream_t stream) {
  (void)in_sizes; (void)n_in; (void)out_size; (void)ws_size;
  const float* x         = (const float*)d_in[0];
  const float* edge_attr = (const float*)d_in[1];
  const float* encW1     = (const float*)d_in[2];
  const float* encb1     = (const float*)d_in[3];
  const float* encW2     = (const float*)d_in[4];
  const float* encb2     = (const float*)d_in[5];
  const float* convW     = (const float*)d_in[6];
  const float* convB     = (const float*)d_in[7];
  const float* mlpW1     = (const float*)d_in[8];
  const float* mlpb1     = (const float*)d_in[9];
  const float* mlpW2     = (const float*)d_in[10];
  const float* mlpb2     = (const float*)d_in[11];
  const float* linW      = (const float*)d_in[12];
  const float* linb      = (const float*)d_in[13];
  const int*   edge_index = (const int*)d_in[14];
  const int*   pick       = (const int*)d_in[15];
  float* out = (float*)d_out;

  // workspace carve-up (floats): ew[4E] | deg4[4N] | h[N*64] | xa[N*64] | xb[N*64]
  float* ws   = (float*)d_ws;
  float* ew   = ws;
  float* deg4 = ew + (size_t)4 * N_EDGES;
  float* h    = deg4 + (size_t)4 * N_NODES;
  float* xa   = h + (size_t)N_NODES * HIDDEN;
  float* xb   = xa + (size_t)N_NODES * HIDDEN;

  const int* row = edge_index;
  const int* col = edge_index + N_EDGES;

  deg_init_kernel<<<(4 * N_NODES + 255) / 256, 256, 0, stream>>>(deg4);
  edge_weights_kernel<<<640, 256, 0, stream>>>(
      edge_attr, encW1, encb1, encW2, encb2, mlpW1, mlpb1, mlpW2, mlpb2, ew);
  deg_accum_kernel<<<(N_EDGES + 255) / 256, 256, 0, stream>>>(col, ew, deg4);
  deg_rsqrt_kernel<<<(4 * N_NODES + 255) / 256, 256, 0, stream>>>(deg4);

  const float* cur = x;
  float* bufs[2] = {xa, xb};
  for (int l = 0; l < 4; ++l) {
    float* xo = bufs[l & 1];
    conv_gemm_kernel<<<391, 256, 0, stream>>>(
        cur, convW + (size_t)l * 64 * 64, convB + (size_t)l * 64,
        deg4 + (size_t)l * N_NODES, h, xo, l > 0 ? 1 : 0);
    scatter_kernel<<<(int)(((long long)N_EDGES * HIDDEN + 255) / 256), 256, 0, stream>>>(
        row, col, ew + (size_t)l * N_EDGES, deg4 + (size_t)l * N_NODES, h, xo);
    cur = xo;
  }
  classify_kernel<<<(N_PICK + 255) / 256, 256, 0, stream>>>(cur, linW, linb, pick, out);
}